// FourierGrid_25890062860522
// MI455X (gfx1250) — compile-verified
//
#include <hip/hip_runtime.h>
#include <math.h>

// FourierGrid trilinear sampler for MI455X (gfx1250).
// Memory-bound random gather: optimize cacheline economy + prefetch distance.

#define CHANNELS 12
#define NUM_FREQ 5
#define PCOMP (1 + 2 * NUM_FREQ)   // 11
#define DD 96
#define DXYZ (DD * DD * DD)        // 884736
#define SX (DD * DD * CHANNELS)    // x stride in transposed layout
#define SY (DD * CHANNELS)         // y stride
#define VARSZ ((size_t)DXYZ * CHANNELS)

struct AxisIdx { int i0, i1; float w; };

__device__ __forceinline__ AxisIdx axis_map(float u) {
    // align_corners=True mapping to [0, DD-1], reference-identical clamping.
    AxisIdx a;
    float f = (u + 1.0f) * (0.5f * (float)(DD - 1));
    int i0 = (int)floorf(f);
    i0 = i0 < 0 ? 0 : (i0 > DD - 1 ? DD - 1 : i0);
    int i1 = i0 + 1; i1 = i1 > DD - 1 ? DD - 1 : i1;
    a.i0 = i0; a.i1 = i1; a.w = f - (float)i0;
    return a;
}

// Per-variant sampling state: pre-multiplied row offsets + lerp weights.
struct VarIdx { int bx0, bx1, by0, by1, bz0, bz1; float wx, wy, wz; };

// ---------------------------------------------------------------------------
// Phase 1: layout transform  grid[(p*12+c)][x][y][z]  ->  tg[p][x][y][z][c]
// Channel-last: one corner fetch = 48 contiguous bytes (3x b128) instead of
// 12 loads scattered across 40 MB. 2D launch (y = variant) avoids int div.
// ---------------------------------------------------------------------------
__global__ __launch_bounds__(256) void fg_transpose_kernel(
    const float* __restrict__ grid, float* __restrict__ tg) {
    int s = blockIdx.x * blockDim.x + threadIdx.x;   // spatial index in [0,DXYZ)
    int p = blockIdx.y;                               // variant
    if (s >= DXYZ) return;
    float v[CHANNELS];
#pragma unroll
    for (int c = 0; c < CHANNELS; ++c)
        v[c] = grid[(size_t)(p * CHANNELS + c) * DXYZ + s];
    float4* dst = (float4*)(tg + (size_t)p * VARSZ + (size_t)s * CHANNELS);
    dst[0] = make_float4(v[0], v[1], v[2], v[3]);
    dst[1] = make_float4(v[4], v[5], v[6], v[7]);
    dst[2] = make_float4(v[8], v[9], v[10], v[11]);
}

// ---------------------------------------------------------------------------
// Transposed-layout gather, software-pipelined:
//   Phase A: compute all 11 embeddings + axis indices, issue all 88 row
//            prefetches (global_prefetch_b8) up front.
//   Phase B: 11 x (8 corners x 3 float4 loads) + trilinear accumulate.
// ---------------------------------------------------------------------------
__global__ __launch_bounds__(256) void fg_gather_t_kernel(
    const float* __restrict__ xyz, const float* __restrict__ tg,
    const float* __restrict__ mn, const float* __restrict__ mx,
    float* __restrict__ out, int N) {
    int n = blockIdx.x * blockDim.x + threadIdx.x;
    if (n >= N) return;
    // Input col0 -> Dx (slow), col1 -> Dy, col2 -> Dz (fast); the [::-1] flip
    // and the grid_sample coord convention cancel.
    float tx = (xyz[3 * n + 0] - mn[0]) / (mx[0] - mn[0]) * 2.0f - 1.0f;
    float ty = (xyz[3 * n + 1] - mn[1]) / (mx[1] - mn[1]) * 2.0f - 1.0f;
    float tz = (xyz[3 * n + 2] - mn[2]) / (mx[2] - mn[2]) * 2.0f - 1.0f;

    // Embedded coordinates for all 11 variants (constant-indexed -> registers).
    float u[PCOMP][3];
    u[0][0] = tx; u[0][1] = ty; u[0][2] = tz;
#pragma unroll
    for (int k = 0; k < NUM_FREQ; ++k) {
        float fr = (float)(1 << k);                  // 2^k, exact
        float s0, c0, s1, c1, s2, c2;
        __sincosf(fr * tx, &s0, &c0);
        __sincosf(fr * ty, &s1, &c1);
        __sincosf(fr * tz, &s2, &c2);
        u[1 + 2 * k][0] = s0; u[2 + 2 * k][0] = c0;
        u[1 + 2 * k][1] = s1; u[2 + 2 * k][1] = c1;
        u[1 + 2 * k][2] = s2; u[2 + 2 * k][2] = c2;
    }

    // Phase A: indices + all prefetches (max distance before dependent loads).
    VarIdx vi[PCOMP];
#pragma unroll
    for (int p = 0; p < PCOMP; ++p) {
        AxisIdx ax = axis_map(u[p][0]);
        AxisIdx ay = axis_map(u[p][1]);
        AxisIdx az = axis_map(u[p][2]);
        vi[p].bx0 = ax.i0 * SX; vi[p].bx1 = ax.i1 * SX;
        vi[p].by0 = ay.i0 * SY; vi[p].by1 = ay.i1 * SY;
        vi[p].bz0 = az.i0 * CHANNELS; vi[p].bz1 = az.i1 * CHANNELS;
        vi[p].wx = ax.w; vi[p].wy = ay.w; vi[p].wz = az.w;
        const float* gp = tg + (size_t)p * VARSZ;
        __builtin_prefetch(gp + vi[p].bx0 + vi[p].by0 + vi[p].bz0, 0, 3);
        __builtin_prefetch(gp + vi[p].bx0 + vi[p].by0 + vi[p].bz1, 0, 3);
        __builtin_prefetch(gp + vi[p].bx0 + vi[p].by1 + vi[p].bz0, 0, 3);
        __builtin_prefetch(gp + vi[p].bx0 + vi[p].by1 + vi[p].bz1, 0, 3);
        __builtin_prefetch(gp + vi[p].bx1 + vi[p].by0 + vi[p].bz0, 0, 3);
        __builtin_prefetch(gp + vi[p].bx1 + vi[p].by0 + vi[p].bz1, 0, 3);
        __builtin_prefetch(gp + vi[p].bx1 + vi[p].by1 + vi[p].bz0, 0, 3);
        __builtin_prefetch(gp + vi[p].bx1 + vi[p].by1 + vi[p].bz1, 0, 3);
    }

    // Phase B: gather + trilinear accumulate.
    float acc[CHANNELS];
#pragma unroll
    for (int c = 0; c < CHANNELS; ++c) acc[c] = 0.0f;

#pragma unroll
    for (int p = 0; p < PCOMP; ++p) {
        const float* gp = tg + (size_t)p * VARSZ;
        int b000 = vi[p].bx0 + vi[p].by0 + vi[p].bz0;
        int b001 = vi[p].bx0 + vi[p].by0 + vi[p].bz1;
        int b010 = vi[p].bx0 + vi[p].by1 + vi[p].bz0;
        int b011 = vi[p].bx0 + vi[p].by1 + vi[p].bz1;
        int b100 = vi[p].bx1 + vi[p].by0 + vi[p].bz0;
        int b101 = vi[p].bx1 + vi[p].by0 + vi[p].bz1;
        int b110 = vi[p].bx1 + vi[p].by1 + vi[p].bz0;
        int b111 = vi[p].bx1 + vi[p].by1 + vi[p].bz1;
        float wx1 = vi[p].wx, wy1 = vi[p].wy, wz1 = vi[p].wz;
        float wx0 = 1.0f - wx1, wy0 = 1.0f - wy1, wz0 = 1.0f - wz1;
        float w000 = wx0 * wy0 * wz0, w001 = wx0 * wy0 * wz1;
        float w010 = wx0 * wy1 * wz0, w011 = wx0 * wy1 * wz1;
        float w100 = wx1 * wy0 * wz0, w101 = wx1 * wy0 * wz1;
        float w110 = wx1 * wy1 * wz0, w111 = wx1 * wy1 * wz1;
#pragma unroll
        for (int j = 0; j < 3; ++j) {
            const float4 v000 = *(const float4*)(gp + b000 + 4 * j);
            const float4 v001 = *(const float4*)(gp + b001 + 4 * j);
            const float4 v010 = *(const float4*)(gp + b010 + 4 * j);
            const float4 v011 = *(const float4*)(gp + b011 + 4 * j);
            const float4 v100 = *(const float4*)(gp + b100 + 4 * j);
            const float4 v101 = *(const float4*)(gp + b101 + 4 * j);
            const float4 v110 = *(const float4*)(gp + b110 + 4 * j);
            const float4 v111 = *(const float4*)(gp + b111 + 4 * j);
            acc[4 * j + 0] += w000 * v000.x + w001 * v001.x + w010 * v010.x + w011 * v011.x
                            + w100 * v100.x + w101 * v101.x + w110 * v110.x + w111 * v111.x;
            acc[4 * j + 1] += w000 * v000.y + w001 * v001.y + w010 * v010.y + w011 * v011.y
                            + w100 * v100.y + w101 * v101.y + w110 * v110.y + w111 * v111.y;
            acc[4 * j + 2] += w000 * v000.z + w001 * v001.z + w010 * v010.z + w011 * v011.z
                            + w100 * v100.z + w101 * v101.z + w110 * v110.z + w111 * v111.z;
            acc[4 * j + 3] += w000 * v000.w + w001 * v001.w + w010 * v010.w + w011 * v011.w
                            + w100 * v100.w + w101 * v101.w + w110 * v110.w + w111 * v111.w;
        }
    }

    const float invP = 1.0f / (float)PCOMP;
    float4* o = (float4*)(out + (size_t)n * CHANNELS);  // 48B-aligned
    o[0] = make_float4(acc[0] * invP, acc[1] * invP, acc[2] * invP, acc[3] * invP);
    o[1] = make_float4(acc[4] * invP, acc[5] * invP, acc[6] * invP, acc[7] * invP);
    o[2] = make_float4(acc[8] * invP, acc[9] * invP, acc[10] * invP, acc[11] * invP);
}

// ---------------------------------------------------------------------------
// Fallback: sample the original layout directly (if d_ws can't hold staging).
// ---------------------------------------------------------------------------
__device__ __forceinline__ void sample_acc_d(const float* __restrict__ gbase,
                                             float ux, float uy, float uz,
                                             float acc[CHANNELS]) {
    AxisIdx ax = axis_map(ux), ay = axis_map(uy), az = axis_map(uz);
    int o000 = (ax.i0 * DD + ay.i0) * DD + az.i0;
    int o001 = (ax.i0 * DD + ay.i0) * DD + az.i1;
    int o010 = (ax.i0 * DD + ay.i1) * DD + az.i0;
    int o011 = (ax.i0 * DD + ay.i1) * DD + az.i1;
    int o100 = (ax.i1 * DD + ay.i0) * DD + az.i0;
    int o101 = (ax.i1 * DD + ay.i0) * DD + az.i1;
    int o110 = (ax.i1 * DD + ay.i1) * DD + az.i0;
    int o111 = (ax.i1 * DD + ay.i1) * DD + az.i1;
    float wx1 = ax.w, wy1 = ay.w, wz1 = az.w;
    float wx0 = 1.0f - wx1, wy0 = 1.0f - wy1, wz0 = 1.0f - wz1;
    float w000 = wx0 * wy0 * wz0, w001 = wx0 * wy0 * wz1;
    float w010 = wx0 * wy1 * wz0, w011 = wx0 * wy1 * wz1;
    float w100 = wx1 * wy0 * wz0, w101 = wx1 * wy0 * wz1;
    float w110 = wx1 * wy1 * wz0, w111 = wx1 * wy1 * wz1;
#pragma unroll
    for (int c = 0; c < CHANNELS; ++c) {
        const float* g = gbase + (size_t)c * DXYZ;
        acc[c] += w000 * g[o000] + w001 * g[o001] + w010 * g[o010] + w011 * g[o011]
                + w100 * g[o100] + w101 * g[o101] + w110 * g[o110] + w111 * g[o111];
    }
}

__global__ __launch_bounds__(256) void fg_gather_d_kernel(
    const float* __restrict__ xyz, const float* __restrict__ grid,
    const float* __restrict__ mn, const float* __restrict__ mx,
    float* __restrict__ out, int N) {
    int n = blockIdx.x * blockDim.x + threadIdx.x;
    if (n >= N) return;
    float tx = (xyz[3 * n + 0] - mn[0]) / (mx[0] - mn[0]) * 2.0f - 1.0f;
    float ty = (xyz[3 * n + 1] - mn[1]) / (mx[1] - mn[1]) * 2.0f - 1.0f;
    float tz = (xyz[3 * n + 2] - mn[2]) / (mx[2] - mn[2]) * 2.0f - 1.0f;

    float acc[CHANNELS];
#pragma unroll
    for (int c = 0; c < CHANNELS; ++c) acc[c] = 0.0f;

    sample_acc_d(grid, tx, ty, tz, acc);
#pragma unroll
    for (int k = 0; k < NUM_FREQ; ++k) {
        float fr = (float)(1 << k);
        float sxv, cxv, syv, cyv, szv, czv;
        __sincosf(fr * tx, &sxv, &cxv);
        __sincosf(fr * ty, &syv, &cyv);
        __sincosf(fr * tz, &szv, &czv);
        const float* gs = grid + (size_t)(1 + 2 * k) * CHANNELS * DXYZ;
        const float* gc = grid + (size_t)(2 + 2 * k) * CHANNELS * DXYZ;
        sample_acc_d(gs, sxv, syv, szv, acc);
        sample_acc_d(gc, cxv, cyv, czv, acc);
    }

    const float invP = 1.0f / (float)PCOMP;
    float4* o = (float4*)(out + (size_t)n * CHANNELS);
    o[0] = make_float4(acc[0] * invP, acc[1] * invP, acc[2] * invP, acc[3] * invP);
    o[1] = make_float4(acc[4] * invP, acc[5] * invP, acc[6] * invP, acc[7] * invP);
    o[2] = make_float4(acc[8] * invP, acc[9] * invP, acc[10] * invP, acc[11] * invP);
}

extern "C" void kernel_launch(void* const* d_in, const int* in_sizes, int n_in,
                              void* d_out, int out_size, void* d_ws, size_t ws_size,
                              hipStream_t stream) {
    (void)n_in; (void)out_size;
    const float* xyz  = (const float*)d_in[0];
    const float* grid = (const float*)d_in[1];
    const float* mn   = (const float*)d_in[2];
    const float* mx   = (const float*)d_in[3];
    float* out = (float*)d_out;
    const int N = in_sizes[0] / 3;

    const size_t staging_bytes = (size_t)PCOMP * VARSZ * sizeof(float);
    if (ws_size >= staging_bytes) {
        float* tg = (float*)d_ws;
        dim3 tgrid((DXYZ + 255) / 256, PCOMP, 1);
        fg_transpose_kernel<<<tgrid, 256, 0, stream>>>(grid, tg);
        fg_gather_t_kernel<<<(N + 255) / 256, 256, 0, stream>>>(xyz, tg, mn, mx, out, N);
    } else {
        fg_gather_d_kernel<<<(N + 255) / 256, 256, 0, stream>>>(xyz, grid, mn, mx, out, N);
    }
}